// SLEGNNLayer_59674275610638
// MI455X (gfx1250) — compile-verified
//
#include <hip/hip_runtime.h>
#include <math.h>

typedef float v2f __attribute__((ext_vector_type(2)));
typedef float v8f __attribute__((ext_vector_type(8)));

#define D 64
#define BN_EPS 1e-5f

__device__ __forceinline__ void atomicAddF(float* p, float v) {
    __hip_atomic_fetch_add(p, v, __ATOMIC_RELAXED, __HIP_MEMORY_SCOPE_AGENT);
}

// ---- degree: dinv buffer doubles as deg accumulator --------------------
__global__ void k_deg_init(float* deg, int n) {
    int i = blockIdx.x * blockDim.x + threadIdx.x;
    if (i < n) deg[i] = 1.0f;  // self-loop contributes 1
}

__global__ void k_deg_count(const int* __restrict__ src, const int* __restrict__ dst,
                            float* deg, int e) {
    int i = blockIdx.x * blockDim.x + threadIdx.x;
    if (i < e) {
        int s = src[i], d = dst[i];
        if (s != d) atomicAddF(&deg[d], 1.0f);
    }
}

__global__ void k_rsqrt_inplace(float* deg, int n) {
    int i = blockIdx.x * blockDim.x + threadIdx.x;
    if (i < n) deg[i] = rsqrtf(deg[i]);  // deg >= 1 always
}

// ---- GEMM h = x @ W via v_wmma_f32_16x16x4_f32 -------------------------
// One wave per 16x16 output tile; K-loop of 16 steps of K=4.
__global__ void k_gemm_wmma(const float* __restrict__ x, const float* __restrict__ W,
                            float* __restrict__ h, int nrows) {
    const int wave = threadIdx.x >> 5;
    const int lane = threadIdx.x & 31;
    const int g = blockIdx.x * (blockDim.x >> 5) + wave;   // tile id (wave-uniform)
    const int mTile = g >> 2;            // D/16 = 4 column tiles
    const int nTile = g & 3;
    const int m0 = mTile << 4;
    const int n0 = nTile << 4;
    if (m0 >= nrows) return;             // wave-uniform: EXEC stays all-1s

    const int half = lane >> 4;          // 0: lanes 0-15, 1: lanes 16-31
    const int l    = lane & 15;

    const float* arow = x + (size_t)(m0 + l) * D;
    v8f c = {};
#pragma unroll
    for (int k0 = 0; k0 < D; k0 += 4) {
        // A fragment: VGPR j holds A[m0+l][k0 + 2*half + j]  -> contiguous float2
        v2f a = *(const v2f*)(arow + k0 + 2 * half);
        // B fragment: VGPR j holds W[k0 + 2*half + j][n0 + l]
        v2f b;
        b.x = W[(k0 + 2 * half + 0) * D + n0 + l];
        b.y = W[(k0 + 2 * half + 1) * D + n0 + l];
        c = __builtin_amdgcn_wmma_f32_16x16x4_f32(false, a, false, b,
                                                  (short)0, c, false, false);
    }
    // C/D: VGPR r holds out[m0 + r + 8*half][n0 + l]
#pragma unroll
    for (int r = 0; r < 8; ++r) {
        h[(size_t)(m0 + r + 8 * half) * D + n0 + l] = c[r];
    }
}

// ---- out = h * dinv^2 (self loop) + bias -------------------------------
__global__ void k_self_init(const float* __restrict__ h, const float* __restrict__ dinv,
                            const float* __restrict__ bias, float* __restrict__ out,
                            size_t total) {
    size_t idx = (size_t)blockIdx.x * blockDim.x + threadIdx.x;
    if (idx < total) {
        size_t i = idx >> 6;
        int    c = (int)(idx & 63);
        float  di = dinv[i];
        out[idx] = h[idx] * di * di + bias[c];
    }
}

// ---- edge scatter: one wave per edge, lane covers 2 channels -----------
__global__ void k_scatter(const int* __restrict__ src, const int* __restrict__ dst,
                          const float* __restrict__ h, const float* __restrict__ dinv,
                          float* __restrict__ out, int e) {
    int wid  = (int)(((size_t)blockIdx.x * blockDim.x + threadIdx.x) >> 5);
    int lane = threadIdx.x & 31;
    if (wid >= e) return;
    int s = src[wid], d = dst[wid];
    if (s == d) return;                      // zero-weight self edge
    float w = dinv[s] * dinv[d];
    v2f hv = *(const v2f*)(h + (size_t)s * D + 2 * lane);
    float* o = out + (size_t)d * D + 2 * lane;
    atomicAddF(o,     hv.x * w);
    atomicAddF(o + 1, hv.y * w);
}

// ---- BN statistics -----------------------------------------------------
__global__ void k_zero_sums(float* sums) {
    if (threadIdx.x < 128) sums[threadIdx.x] = 0.0f;
}

__global__ void k_stats(const float* __restrict__ out, float* __restrict__ sums, int n) {
    __shared__ float p1[256];
    __shared__ float p2[256];
    int c  = threadIdx.x & 63;
    int rl = threadIdx.x >> 6;               // 0..3
    float s = 0.0f, q = 0.0f;
    for (int r = blockIdx.x * 4 + rl; r < n; r += gridDim.x * 4) {
        float v = out[(size_t)r * D + c];
        s += v;
        q += v * v;
    }
    p1[threadIdx.x] = s;
    p2[threadIdx.x] = q;
    __syncthreads();
    if (threadIdx.x < 64) {
        int t = threadIdx.x;
        float S = p1[t] + p1[t + 64] + p1[t + 128] + p1[t + 192];
        float Q = p2[t] + p2[t + 64] + p2[t + 128] + p2[t + 192];
        atomicAddF(&sums[t],      S);
        atomicAddF(&sums[64 + t], Q);
    }
}

__global__ void k_bn_params(const float* __restrict__ sums,
                            const float* __restrict__ bnw, const float* __restrict__ bnb,
                            float* __restrict__ scale, float* __restrict__ shift, int n) {
    int c = threadIdx.x;
    if (c < 64) {
        float inv_n = 1.0f / (float)n;
        float mean = sums[c] * inv_n;
        float var  = sums[64 + c] * inv_n - mean * mean;   // biased var
        float sc   = bnw[c] * rsqrtf(var + BN_EPS);
        scale[c] = sc;
        shift[c] = bnb[c] - mean * sc;
    }
}

// ---- y = relu(out*scale + shift), in place on d_out --------------------
__global__ void k_bn_relu(float* __restrict__ out, const float* __restrict__ scale,
                          const float* __restrict__ shift, size_t total) {
    size_t idx = (size_t)blockIdx.x * blockDim.x + threadIdx.x;
    if (idx < total) {
        int c = (int)(idx & 63);
        float v = out[idx] * scale[c] + shift[c];
        out[idx] = v > 0.0f ? v : 0.0f;
    }
}

extern "C" void kernel_launch(void* const* d_in, const int* in_sizes, int n_in,
                              void* d_out, int out_size, void* d_ws, size_t ws_size,
                              hipStream_t stream) {
    const float* x    = (const float*)d_in[0];
    const int*   eidx = (const int*)d_in[1];
    const float* W    = (const float*)d_in[3];
    const float* bias = (const float*)d_in[4];
    const float* bnw  = (const float*)d_in[5];
    const float* bnb  = (const float*)d_in[6];

    const int N = in_sizes[0] / D;          // 100000
    const int E = in_sizes[1] / 2;          // 1600000
    const int* src = eidx;
    const int* dst = eidx + E;

    // workspace layout (floats): [dinv: N][h: N*D][sums: 128][scale: 64][shift: 64]
    float* ws    = (float*)d_ws;
    float* dinv  = ws;
    float* h     = ws + N;
    float* sums  = h + (size_t)N * D;
    float* scale = sums + 128;
    float* shift = scale + 64;

    float* out = (float*)d_out;             // N*D accumulator, finalized in place
    const size_t total = (size_t)N * D;

    // 1) degrees -> dinv
    k_deg_init<<<(N + 255) / 256, 256, 0, stream>>>(dinv, N);
    k_deg_count<<<(E + 255) / 256, 256, 0, stream>>>(src, dst, dinv, E);
    k_rsqrt_inplace<<<(N + 255) / 256, 256, 0, stream>>>(dinv, N);

    // 2) h = x @ W  (WMMA, one wave per 16x16 tile, 8 waves per block)
    {
        int tiles  = ((N + 15) / 16) * (D / 16);
        int blocks = (tiles + 7) / 8;
        k_gemm_wmma<<<blocks, 256, 0, stream>>>(x, W, h, N);
    }

    // 3) out = self-loop term + bias, then scatter edges
    k_self_init<<<(int)((total + 255) / 256), 256, 0, stream>>>(h, dinv, bias, out, total);
    k_scatter<<<(E + 7) / 8, 256, 0, stream>>>(src, dst, h, dinv, out, E);

    // 4) BatchNorm stats + affine + ReLU
    k_zero_sums<<<1, 128, 0, stream>>>(sums);
    k_stats<<<1024, 256, 0, stream>>>(out, sums, N);
    k_bn_params<<<1, 64, 0, stream>>>(sums, bnw, bnb, scale, shift, N);
    k_bn_relu<<<(int)((total + 255) / 256), 256, 0, stream>>>(out, scale, shift, total);
}